// DGIBNN_24034636989228
// MI455X (gfx1250) — compile-verified
//
#include <hip/hip_runtime.h>
#include <math.h>

// ---------------------------------------------------------------------------
// DGIB temporal GAT for MI455X (gfx1250, wave32).
//   N=30000 nodes, E=510000 edges/snapshot, T=2, F_IN=H1=128, OUT=64.
// GEMMs use V_WMMA_F32_16X16X4_F32 (fp32 matrix pipe, keeps fp32 precision).
// Edge attention logits are factored into per-node scalars (saves ~16x gather).
// ---------------------------------------------------------------------------

#define NN      30000
#define TT      2
#define EE      510000
#define F_IN    128
#define H1C     128
#define OUTC    64
#define AGG     0.8f
#define NEG_SL  0.2f

typedef __attribute__((ext_vector_type(2))) float v2f;
typedef __attribute__((ext_vector_type(8))) float v8f;

__device__ __forceinline__ float wave_sum(float v) {
    #pragma unroll
    for (int off = 16; off > 0; off >>= 1) v += __shfl_xor(v, off, 32);
    return v;
}

// --------------------------- utility kernels -------------------------------

__global__ void k_zero(float* __restrict__ p, long n) {
    long i = (long)blockIdx.x * blockDim.x + threadIdx.x;
    long stride = (long)gridDim.x * blockDim.x;
    for (; i < n; i += stride) p[i] = 0.0f;
}

__global__ void k_relu(float* __restrict__ p, long n) {
    long i = (long)blockIdx.x * blockDim.x + threadIdx.x;
    long stride = (long)gridDim.x * blockDim.x;
    for (; i < n; i += stride) p[i] = fmaxf(p[i], 0.0f);
}

// out_t1 = 0.4*out_t0 + 0.2*out_t1   (temporal fusion, T=2, sigmoid(0)=0.5)
__global__ void k_fuse(float* __restrict__ t1, const float* __restrict__ t0, long n) {
    long i = (long)blockIdx.x * blockDim.x + threadIdx.x;
    long stride = (long)gridDim.x * blockDim.x;
    for (; i < n; i += stride) t1[i] = 0.4f * t0[i] + 0.2f * t1[i];
}

// --------------------------- WMMA fp32 GEMM --------------------------------
// H[M,Cout] = X[M,K] @ W[K,Cout].  One wave computes one 16x16 tile of H via
// V_WMMA_F32_16X16X4_F32 over K in steps of 4.  Grid: (M/16, Cout/128),
// block 256 = 8 waves -> 8 column tiles per block.
__global__ void k_gemm_wmma(const float* __restrict__ X, const float* __restrict__ W,
                            float* __restrict__ H, int M, int K, int Cout) {
    const int wave = threadIdx.x >> 5;
    const int lane = threadIdx.x & 31;
    const int half = lane >> 4;       // 0: lanes 0-15, 1: lanes 16-31
    const int lm   = lane & 15;
    const int m0 = blockIdx.x * 16;
    const int c0 = (blockIdx.y * 8 + wave) * 16;
    if (m0 >= M || c0 >= Cout) return;           // wave-uniform guard

    v8f acc = {};
    const float* xrow = X + (size_t)(m0 + lm) * K;
    for (int k0 = 0; k0 < K; k0 += 4) {
        const int ka = k0 + 2 * half;
        // A 16x4: VGPR0 = K{0|2}, VGPR1 = K{1|3} per lane half
        v2f a;
        a.x = xrow[ka];
        a.y = xrow[ka + 1];
        // B 4x16: same K split across lane halves, N = lm across lanes
        v2f b;
        b.x = W[(size_t)ka * Cout + c0 + lm];
        b.y = W[(size_t)(ka + 1) * Cout + c0 + lm];
        acc = __builtin_amdgcn_wmma_f32_16x16x4_f32(
            /*neg_a=*/false, a, /*neg_b=*/false, b,
            /*c_mod=*/(short)0, acc, /*reuse_a=*/false, /*reuse_b=*/false);
    }
    // C/D layout: VGPR r -> row m0 + r + 8*half, col c0 + lm
    float* hrow = H + (size_t)(m0 + 8 * half) * Cout + c0 + lm;
    #pragma unroll
    for (int r = 0; r < 8; ++r) hrow[(size_t)r * Cout] = acc[r];
}

// ---------------------- per-node attention scalars -------------------------
// s_dst[n] = dot(h[n], att[0:2C]);  s_src[n] = dot(h[n], att[2C:4C])
__global__ void k_node_scores(const float* __restrict__ H, const float* __restrict__ att,
                              float* __restrict__ s_dst, float* __restrict__ s_src,
                              int n_nodes, int C2) {
    const int wid  = (int)((blockIdx.x * (long)blockDim.x + threadIdx.x) >> 5);
    const int lane = threadIdx.x & 31;
    if (wid >= n_nodes) return;
    const float* h = H + (size_t)wid * C2;
    float a0 = 0.0f, a1 = 0.0f;
    for (int c = lane; c < C2; c += 32) {
        const float hv = h[c];
        a0 += hv * att[c];
        a1 += hv * att[C2 + c];
    }
    a0 = wave_sum(a0);
    a1 = wave_sum(a1);
    if (lane == 0) { s_dst[wid] = a0; s_src[wid] = a1; }
}

// ------------------------------ degree -------------------------------------
__global__ void k_degree(const int* __restrict__ dst, float* __restrict__ deg, int E) {
    int e = blockIdx.x * blockDim.x + threadIdx.x;
    if (e < E) atomicAdd(&deg[dst[e]], 1.0f);
}

// ----------------------- edge attention + scatter-add ----------------------
// One wave32 per edge: uniform logit/p/alpha, lanes scatter consecutive
// channels (coalesced L2 atomics).  Lane 0 accumulates the structure-KL term.
__global__ void k_edge_aggr(const float* __restrict__ H,
                            const int* __restrict__ src, const int* __restrict__ dst,
                            const float* __restrict__ s_dst, const float* __restrict__ s_src,
                            const float* __restrict__ deg,
                            float* __restrict__ aggr, float* __restrict__ skl_acc,
                            int E, int C2) {
    const int e    = (int)((blockIdx.x * (long)blockDim.x + threadIdx.x) >> 5);
    const int lane = threadIdx.x & 31;
    if (e >= E) return;
    const int s = src[e];
    const int d = dst[e];
    float logit = s_dst[d] + s_src[s];
    logit = (logit > 0.0f) ? logit : NEG_SL * logit;                 // leaky_relu
    float p = 1.0f / (1.0f + expf(-logit));
    p = fminf(0.99f, fmaxf(0.01f, p));                               // clip
    const float alpha = p / deg[d];

    const float* hs = H + (size_t)s * C2;
    float* ag = aggr + (size_t)d * C2;
    for (int c = lane; c < C2; c += 32)
        atomicAdd(&ag[c], hs[c] * alpha);

    if (lane == 0) {
        const float q = 1.0f / (1.0f + expf(-1.0f / 15.0f));         // sigmoid(1/NBSZ)
        const float skl = p * logf(p / q) + (1.0f - p) * logf((1.0f - p) / (1.0f - q));
        atomicAdd(skl_acc, skl);
    }
}

// ----------------- bias + reparam split + KL + mu output -------------------
// mu = (aggr+b)[:C];  std = softplus((aggr+b)[C:]) + 1e-10
// kl = -log(std) + 0.5*(std^2 + mu^2) - 0.5 ; ixz_acc += sum_C kl
__global__ void k_node_finalize(const float* __restrict__ aggr, const float* __restrict__ bias,
                                float* __restrict__ mu_out, float* __restrict__ ixz_acc,
                                int n_nodes, int C) {
    const int wid  = (int)((blockIdx.x * (long)blockDim.x + threadIdx.x) >> 5);
    const int lane = threadIdx.x & 31;
    if (wid >= n_nodes) return;
    const float* ag = aggr + (size_t)wid * (2 * C);
    float klsum = 0.0f;
    for (int c = lane; c < C; c += 32) {
        const float mu = ag[c] + bias[c];
        const float xr = ag[C + c] + bias[C + c];
        const float sp = (xr > 20.0f) ? xr : log1pf(expf(xr));       // softplus
        const float sd = sp + 1e-10f;
        klsum += -logf(sd) + 0.5f * (sd * sd + mu * mu) - 0.5f;
        mu_out[(size_t)wid * C + c] = mu;
    }
    klsum = wave_sum(klsum);
    if (lane == 0) atomicAdd(ixz_acc, klsum);
}

// accum = [ixz1_sum, skl1_sum, ixz2_sum, skl2_sum]
__global__ void k_scalars(const float* __restrict__ accum, float* __restrict__ out2) {
    if (threadIdx.x == 0 && blockIdx.x == 0) {
        const float inv_nt = 1.0f / ((float)NN * (float)TT);
        const float inv_et = 1.0f / ((float)EE * (float)TT);
        out2[0] = 0.5f * (accum[0] + accum[2]) * inv_nt;   // ixz_mean
        out2[1] = 0.5f * (accum[1] + accum[3]) * inv_et;   // skl_mean
    }
}

// ---------------------------------------------------------------------------

extern "C" void kernel_launch(void* const* d_in, const int* in_sizes, int n_in,
                              void* d_out, int out_size, void* d_ws, size_t ws_size,
                              hipStream_t stream) {
    const float* x_all = (const float*)d_in[0];   // [T, N, 128]
    const int*   ei    = (const int*)  d_in[1];   // [T, 2, E]
    const float* W1    = (const float*)d_in[2];   // [128, 256]
    const float* att1  = (const float*)d_in[3];   // [512]
    const float* b1    = (const float*)d_in[4];   // [256]
    const float* W2    = (const float*)d_in[5];   // [128, 128]
    const float* att2  = (const float*)d_in[6];   // [256]
    const float* b2    = (const float*)d_in[7];   // [128]
    float* out = (float*)d_out;                   // [T*N*64] ++ [ixz, skl]

    // workspace layout (floats)
    float* ws    = (float*)d_ws;
    float* hbuf  = ws;                                   // N*256
    float* aggr  = hbuf + (size_t)NN * 256;              // N*256
    float* mu1   = aggr + (size_t)NN * 256;              // T*N*128
    float* s_d   = mu1  + (size_t)TT * NN * 128;         // N
    float* s_s   = s_d + NN;                             // N
    float* degv  = s_s + NN;                             // N
    float* accum = degv + NN;                            // 4

    const int BLK = 256;
    const int gNodeWave = (NN * 32 + BLK - 1) / BLK;     // wave-per-node grids
    const int gEdgeWave = (EE * 32 + BLK - 1) / BLK;     // wave-per-edge grid
    const int gEdge     = (EE + BLK - 1) / BLK;

    k_zero<<<1024, BLK, 0, stream>>>(accum, 4);

    // ---------------- Layer 1: F_IN=128 -> C=128 (2C=256) ----------------
    for (int t = 0; t < TT; ++t) {
        const float* xin  = x_all + (size_t)t * NN * F_IN;
        const int* e_src  = ei + (size_t)t * 2 * EE;
        const int* e_dst  = e_src + EE;

        k_gemm_wmma<<<dim3(NN / 16, 256 / 128), BLK, 0, stream>>>(xin, W1, hbuf, NN, F_IN, 256);
        k_zero<<<4096, BLK, 0, stream>>>(aggr, (long)NN * 256);
        k_zero<<<1024, BLK, 0, stream>>>(degv, NN);
        k_node_scores<<<gNodeWave, BLK, 0, stream>>>(hbuf, att1, s_d, s_s, NN, 256);
        k_degree<<<gEdge, BLK, 0, stream>>>(e_dst, degv, EE);
        k_edge_aggr<<<gEdgeWave, BLK, 0, stream>>>(hbuf, e_src, e_dst, s_d, s_s, degv,
                                                   aggr, &accum[1], EE, 256);
        k_node_finalize<<<gNodeWave, BLK, 0, stream>>>(aggr, b1, mu1 + (size_t)t * NN * H1C,
                                                       &accum[0], NN, H1C);
    }
    k_fuse<<<2048, BLK, 0, stream>>>(mu1 + (size_t)NN * H1C, mu1, (long)NN * H1C);
    k_relu<<<4096, BLK, 0, stream>>>(mu1, (long)TT * NN * H1C);

    // ---------------- Layer 2: H1=128 -> C=64 (2C=128) ----------------
    for (int t = 0; t < TT; ++t) {
        const float* xin  = mu1 + (size_t)t * NN * H1C;
        const int* e_src  = ei + (size_t)t * 2 * EE;
        const int* e_dst  = e_src + EE;

        k_gemm_wmma<<<dim3(NN / 16, 128 / 128), BLK, 0, stream>>>(xin, W2, hbuf, NN, H1C, 128);
        k_zero<<<4096, BLK, 0, stream>>>(aggr, (long)NN * 128);
        k_zero<<<1024, BLK, 0, stream>>>(degv, NN);
        k_node_scores<<<gNodeWave, BLK, 0, stream>>>(hbuf, att2, s_d, s_s, NN, 128);
        k_degree<<<gEdge, BLK, 0, stream>>>(e_dst, degv, EE);
        k_edge_aggr<<<gEdgeWave, BLK, 0, stream>>>(hbuf, e_src, e_dst, s_d, s_s, degv,
                                                   aggr, &accum[3], EE, 128);
        k_node_finalize<<<gNodeWave, BLK, 0, stream>>>(aggr, b2, out + (size_t)t * NN * OUTC,
                                                       &accum[2], NN, OUTC);
    }
    k_fuse<<<2048, BLK, 0, stream>>>(out + (size_t)NN * OUTC, out, (long)NN * OUTC);
    k_scalars<<<1, 32, 0, stream>>>(accum, out + (size_t)TT * NN * OUTC);
}